// OneByOneMergeTwo_20538533609783
// MI455X (gfx1250) — compile-verified
//
#include <hip/hip_runtime.h>
#include <hip/hip_bf16.h>

// relu(w0*t1 + w1*t2 + b)  -- pure HBM-bandwidth-bound streaming kernel.
// 308 MB moved per call, ~77 MFLOP => AI ~ 0.25 FLOP/B. At 23.3 TB/s the
// floor is ~13 us; optimize only data movement:
//   * 128-bit vector loads/stores (512 B per wave32 per instruction)
//   * non-temporal (TH=NT) policy: 308 MB working set > 192 MB L2, touched once
//   * 4 independent float4 iterations per thread for memory-level parallelism

typedef float v4f __attribute__((ext_vector_type(4)));

#define ELEMS_PER_THREAD 4   // float4s per thread
#define BLOCK 256            // 8 wave32s per block

__global__ __launch_bounds__(BLOCK) void
OneByOneMergeTwo_kernel(const v4f* __restrict__ t1,
                        const v4f* __restrict__ t2,
                        const float* __restrict__ w,
                        const float* __restrict__ b,
                        v4f* __restrict__ out,
                        long long n4)
{
    // Uniform scalar parameters -> scalar loads (KMcnt path), broadcast to wave.
    const float w0 = w[0];
    const float w1 = w[1];
    const float bb = b[0];

    const long long base =
        (long long)blockIdx.x * (BLOCK * ELEMS_PER_THREAD) + threadIdx.x;

#pragma unroll
    for (int e = 0; e < ELEMS_PER_THREAD; ++e) {
        const long long idx = base + (long long)e * BLOCK;  // coalesced per iter
        if (idx < n4) {
            // Non-temporal 128-bit loads: stream past L2 (data is read once).
            v4f a = __builtin_nontemporal_load(&t1[idx]);
            v4f c = __builtin_nontemporal_load(&t2[idx]);

            v4f r;
            r.x = fmaxf(fmaf(w0, a.x, fmaf(w1, c.x, bb)), 0.0f);
            r.y = fmaxf(fmaf(w0, a.y, fmaf(w1, c.y, bb)), 0.0f);
            r.z = fmaxf(fmaf(w0, a.z, fmaf(w1, c.z, bb)), 0.0f);
            r.w = fmaxf(fmaf(w0, a.w, fmaf(w1, c.w, bb)), 0.0f);

            // Non-temporal 128-bit store: avoid write-allocate pollution.
            __builtin_nontemporal_store(r, &out[idx]);
        }
    }
}

// Scalar tail (out_size not divisible by 4 -- not hit for this shape, but safe).
__global__ void
OneByOneMergeTwo_tail(const float* __restrict__ t1,
                      const float* __restrict__ t2,
                      const float* __restrict__ w,
                      const float* __restrict__ b,
                      float* __restrict__ out,
                      long long start, long long n)
{
    long long i = start + blockIdx.x * blockDim.x + threadIdx.x;
    if (i < n) {
        float r = fmaf(w[0], t1[i], fmaf(w[1], t2[i], b[0]));
        out[i] = fmaxf(r, 0.0f);
    }
}

extern "C" void kernel_launch(void* const* d_in, const int* in_sizes, int n_in,
                              void* d_out, int out_size, void* d_ws, size_t ws_size,
                              hipStream_t stream)
{
    const float* t1 = (const float*)d_in[0];
    const float* t2 = (const float*)d_in[1];
    const float* w  = (const float*)d_in[2];
    const float* b  = (const float*)d_in[3];
    float* out = (float*)d_out;

    const long long n  = (long long)out_size;      // 25,690,112 for this shape
    const long long n4 = n / 4;                    // float4 count
    const long long tail_start = n4 * 4;

    if (n4 > 0) {
        const long long per_block = (long long)BLOCK * ELEMS_PER_THREAD;
        const long long grid = (n4 + per_block - 1) / per_block;
        OneByOneMergeTwo_kernel<<<(unsigned)grid, BLOCK, 0, stream>>>(
            (const v4f*)t1, (const v4f*)t2, w, b, (v4f*)out, n4);
    }
    if (tail_start < n) {
        const long long tn = n - tail_start;
        const long long grid = (tn + 255) / 256;
        OneByOneMergeTwo_tail<<<(unsigned)grid, 256, 0, stream>>>(
            t1, t2, w, b, out, tail_start, n);
    }
}